// UncertaintyEstimatorIOU_34600256537503
// MI455X (gfx1250) — compile-verified
//
#include <hip/hip_runtime.h>

#define IOU_THR 0.5f
#define IOU_EPS 1e-7f

#define NS      16                    // dropout samples
#define KBOX    1024                  // boxes per sample
#define ROWS_PB 16                    // pred rows per block
#define BLK     256                   // ROWS_PB * NS threads
// Padded slice stride: 4 floats of skew per slice so the 16 lanes scanning the
// same k at different j hit 16 distinct LDS banks (16 KB stride would alias to
// the same bank for every j).
#define SLICE_F (KBOX * 4 + 4)
#define LDS_BYTES (NS * SLICE_F * 4)  // 262,400 B — needs CDNA5's 320 KB LDS

// Grid: nPred/ROWS_PB blocks x 256 threads. Thread (tid>>4, tid&15) owns
// (pred_row_local, sample_j): scans sample j's 1024 boxes for the first
// IoU > 0.5, then a 16-lane shuffle tree reduces over j.
__global__ __launch_bounds__(BLK) void UncertaintyEstimatorIOU_kernel(
    const float* __restrict__ pred,      // (nPred, 6) xywh in cols 0..3
    const float* __restrict__ dropout,   // (NS, KBOX, 6) xywh in cols 0..3
    float* __restrict__ out)             // (nPred,)
{
    extern __shared__ float smem[];      // NS skewed slices of KBOX xyxy boxes

    const int tid = threadIdx.x;
    // Low 32 bits of a generic pointer to LDS = the LDS byte offset.
    const unsigned lds_base = (unsigned)(unsigned long long)(void*)smem;

    // ---- Bulk stage: all NS*KBOX boxes (16B of each 24B record) via the
    // CDNA5 async global->LDS path (ASYNCcnt; HW throttles at 63 in flight).
    for (int p = 0; p < (NS * KBOX) / BLK; ++p) {
        const int e  = tid + p * BLK;           // flat (j,k) index
        const int jj = e >> 10;
        const int kk = e & (KBOX - 1);
        const unsigned voff = (unsigned)(e * 24);                        // global bytes
        const unsigned doff = lds_base + (unsigned)((jj * SLICE_F + kk * 4) * 4);
        asm volatile("global_load_async_to_lds_b128 %0, %1, %2"
                     :: "v"(doff), "v"(voff), "s"(dropout)
                     : "memory");
    }

    // ---- Overlap: load this thread's pred box while the DMA is in flight.
    const int row = blockIdx.x * ROWS_PB + (tid >> 4);
    const int jmy = tid & (NS - 1);
    const float2 pxy = *(const float2*)(pred + (size_t)row * 6);
    const float2 pwh = *(const float2*)(pred + (size_t)row * 6 + 2);
    const float ax1 = pxy.x - 0.5f * pwh.x;
    const float ay1 = pxy.y - 0.5f * pwh.y;
    const float ax2 = pxy.x + 0.5f * pwh.x;
    const float ay2 = pxy.y + 0.5f * pwh.y;
    const float areaA = (ax2 - ax1) * (ay2 - ay1);

    asm volatile("s_wait_asynccnt 0" ::: "memory");
    __syncthreads();

    // ---- Convert staged xywh -> xyxy in place (each thread owns 64 slots).
    for (int p = 0; p < (NS * KBOX) / BLK; ++p) {
        const int e  = tid + p * BLK;
        const int jj = e >> 10;
        const int kk = e & (KBOX - 1);
        float4* ptr = (float4*)(&smem[jj * SLICE_F + kk * 4]);
        const float4 v = *ptr;
        const float hx = 0.5f * v.z, hy = 0.5f * v.w;
        *ptr = make_float4(v.x - hx, v.y - hy, v.x + hx, v.y + hy);
    }
    __syncthreads();

    // ---- Early-exit scan of my sample slice: first box with IoU > 0.5.
    const float* slice = &smem[jmy * SLICE_F];
    float sel = 0.0f;
    int   fnd = 0;
    for (int k = 0; k < KBOX; ++k) {
        const float4 b = *(const float4*)(slice + k * 4);
        float w = fminf(ax2, b.z) - fmaxf(ax1, b.x);
        float h = fminf(ay2, b.w) - fmaxf(ay1, b.y);
        w = fmaxf(w, 0.0f);
        h = fmaxf(h, 0.0f);
        const float inter = w * h;
        const float areaB = (b.z - b.x) * (b.w - b.y);
        const float iou = inter / (areaA + areaB - inter + IOU_EPS);
        if (iou > IOU_THR) {            // argmax-of-mask semantics
            sel = iou;
            fnd = 1;
            break;
        }
    }

    // ---- Reduce (sum sel, count found) over the 16 j-lanes of this row.
    // xor masks < 16 stay inside each 16-lane half of the wave32.
    float sum = sel;
    int   cnt = fnd;
#pragma unroll
    for (int m = 1; m < NS; m <<= 1) {
        sum += __shfl_xor(sum, m, 32);
        cnt += __shfl_xor(cnt, m, 32);
    }
    if (jmy == 0) {
        const float unc = sum / (float)(cnt > 0 ? cnt : 1);
        out[row] = (unc - IOU_THR) * 2.0f;   // (unc - 0.5) / (1 - 0.5)
    }
}

extern "C" void kernel_launch(void* const* d_in, const int* in_sizes, int n_in,
                              void* d_out, int out_size, void* d_ws, size_t ws_size,
                              hipStream_t stream) {
    (void)n_in; (void)d_ws; (void)ws_size; (void)out_size;
    const float* pred    = (const float*)d_in[0];   // (2048, 6) fp32
    const float* dropout = (const float*)d_in[1];   // (16, 1024, 6) fp32
    // d_in[2] (dropout_cls_confs) is unused by the reference.
    float* out = (float*)d_out;                     // (2048,) fp32

    const int nPred = in_sizes[0] / 6;              // 2048
    UncertaintyEstimatorIOU_kernel<<<nPred / ROWS_PB, BLK, LDS_BYTES, stream>>>(
        pred, dropout, out);
}